// GnnBasic_39908836114582
// MI455X (gfx1250) — compile-verified
//
#include <hip/hip_runtime.h>

typedef __attribute__((ext_vector_type(2))) float v2f;
typedef __attribute__((ext_vector_type(8))) float v8f;

// ---------------------------------------------------------------------------
// K1: deg[v] = 1 (self-loop).  deg lives (as u32) in the buffer that later
// holds dinv (f32) — same region, converted in-place by K3.
// ---------------------------------------------------------------------------
__global__ void k_init_deg(unsigned* __restrict__ deg, int N) {
    int v = blockIdx.x * blockDim.x + threadIdx.x;
    if (v < N) deg[v] = 1u;
}

// K2 (vectorized): deg[dst[e]] += 1, 4 edges/thread via b128 index loads.
__global__ void k_count_deg4(const int4* __restrict__ dst4,
                             unsigned* __restrict__ deg, int E4) {
    int i = blockIdx.x * blockDim.x + threadIdx.x;
    if (i < E4) {
        int4 d = dst4[i];
        atomicAdd(&deg[d.x], 1u);
        atomicAdd(&deg[d.y], 1u);
        atomicAdd(&deg[d.z], 1u);
        atomicAdd(&deg[d.w], 1u);
    }
}

// K2 (scalar fallback / tail).
__global__ void k_count_deg(const int* __restrict__ dst,
                            unsigned* __restrict__ deg, int E) {
    int e = blockIdx.x * blockDim.x + threadIdx.x;
    if (e < E) atomicAdd(&deg[dst[e]], 1u);
}

// K3: dinv[v] = rsqrt(deg[v]) (in place over the u32 counts) and seed the
// layer-1 accumulator with the self-loop term: A[v] = dinv[v]^2 * x[v,0].
__global__ void k_dinv_selfloop(unsigned* __restrict__ deg_u,
                                float* __restrict__ dinv,
                                const float* __restrict__ x,
                                float* __restrict__ A, int N) {
    int v = blockIdx.x * blockDim.x + threadIdx.x;
    if (v < N) {
        float d  = (float)deg_u[v];          // >= 1 always (self-loop)
        float dv = rsqrtf(d);
        dinv[v]  = dv;                       // overwrite count with dinv
        A[v]     = dv * dv * x[4 * v];       // x is (N,4); column 0
    }
}

// K4 / K7 (vectorized): acc[dst] += dinv[src]*dinv[dst]*val[src*stride],
// 4 edges/thread via b128 index loads.  stride=4 walks column 0 of x.
__global__ void k_scatter4(const int4* __restrict__ src4,
                           const int4* __restrict__ dst4,
                           const float* __restrict__ dinv,
                           const float* __restrict__ val, int stride,
                           float* __restrict__ acc, int E4) {
    int i = blockIdx.x * blockDim.x + threadIdx.x;
    if (i < E4) {
        int4 s = src4[i];
        int4 d = dst4[i];
        float m0 = dinv[s.x] * dinv[d.x] * val[s.x * stride];
        float m1 = dinv[s.y] * dinv[d.y] * val[s.y * stride];
        float m2 = dinv[s.z] * dinv[d.z] * val[s.z * stride];
        float m3 = dinv[s.w] * dinv[d.w] * val[s.w * stride];
        atomicAdd(&acc[d.x], m0);
        atomicAdd(&acc[d.y], m1);
        atomicAdd(&acc[d.z], m2);
        atomicAdd(&acc[d.w], m3);
    }
}

// K4 / K7 (scalar fallback / tail).
__global__ void k_scatter(const int* __restrict__ src,
                          const int* __restrict__ dst,
                          const float* __restrict__ dinv,
                          const float* __restrict__ val, int stride,
                          float* __restrict__ acc, int E) {
    int e = blockIdx.x * blockDim.x + threadIdx.x;
    if (e < E) {
        int s = src[e];
        int d = dst[e];
        atomicAdd(&acc[d], dinv[s] * dinv[d] * val[s * stride]);
    }
}

// ---------------------------------------------------------------------------
// K5: per-node MLP via V_WMMA_F32_16X16X4_F32.  Each wave32 handles 16 nodes.
//   D[f, node] = W1[f] * A[node] + b1[f]       (one WMMA, K=1 outer product)
//   g[node]    = sum_f relu(D[f,node]) * W2[f] (8 FMAs/lane + lane-16 fold)
// Operand placement (wave32, 16x16x4 f32):
//   A-op (16x4, M=feature): lanes 0-15, VGPR0 = W1[lane];       all else 0
//   B-op (4x16, N=node):    lanes 0-15, VGPR0 = A[base+lane];   all else 0
//   C/D  (16x16):  VGPR j: lanes 0-15 -> M=j, N=lane ; lanes 16-31 -> M=j+8
// EXEC must be all-1s for WMMA: all lanes stay active, loads clamped,
// stores guarded.
// ---------------------------------------------------------------------------
__global__ void k_transform_wmma(const float* __restrict__ A,
                                 const float* __restrict__ W1,
                                 const float* __restrict__ b1,
                                 const float* __restrict__ W2,
                                 float* __restrict__ g, int N) {
    int lane   = threadIdx.x & 31;
    int wave   = threadIdx.x >> 5;
    int lane16 = lane & 15;
    bool lo    = lane < 16;
    int base   = (blockIdx.x * (blockDim.x >> 5) + wave) * 16;
    int node   = base + lane16;
    int cnode  = node < N ? node : (N - 1);

    v2f amat;
    amat.x = lo ? W1[lane16] : 0.0f;
    amat.y = 0.0f;

    v2f bmat;
    bmat.x = lo ? A[cnode] : 0.0f;
    bmat.y = 0.0f;

    int fbase = lo ? 0 : 8;
    v8f c;
#pragma unroll
    for (int j = 0; j < 8; ++j) c[j] = b1[fbase + j];

    // D = A x B + C   -> v_wmma_f32_16x16x4_f32
    v8f d = __builtin_amdgcn_wmma_f32_16x16x4_f32(
        /*neg_a=*/false, amat, /*neg_b=*/false, bmat,
        /*c_mod=*/(short)0, c, /*reuse_a=*/false, /*reuse_b=*/false);

    float partial = 0.0f;
#pragma unroll
    for (int j = 0; j < 8; ++j) {
        float h = d[j];
        h = h > 0.0f ? h : 0.0f;             // ReLU
        partial = fmaf(h, W2[fbase + j], partial);
    }
    // features 0-7 live in lane L, features 8-15 in lane L+16: fold halves.
    partial += __shfl_xor(partial, 16, 32);

    if (lo && node < N) g[node] = partial;
}

// K6: seed output with self-loop + bias: out[v] = dinv[v]^2 * g[v] + b2.
__global__ void k_out_init(const float* __restrict__ dinv,
                           const float* __restrict__ gv,
                           const float* __restrict__ b2,
                           float* __restrict__ out, int N) {
    int v = blockIdx.x * blockDim.x + threadIdx.x;
    if (v < N) {
        float dv = dinv[v];
        out[v] = dv * dv * gv[v] + b2[0];
    }
}

extern "C" void kernel_launch(void* const* d_in, const int* in_sizes, int n_in,
                              void* d_out, int out_size, void* d_ws, size_t ws_size,
                              hipStream_t stream) {
    const int N = in_sizes[0] / 4;       // x is (N,4)
    const int E = in_sizes[5] / 2;       // edge_index is (2,E)

    const float* x  = (const float*)d_in[0];
    const float* W1 = (const float*)d_in[1];
    const float* b1 = (const float*)d_in[2];
    const float* W2 = (const float*)d_in[3];
    const float* b2 = (const float*)d_in[4];
    const int*   src = (const int*)d_in[5];
    const int*   dst = src + E;

    // Workspace: [dinv (u32 deg then f32) | A | g], 3*N floats = 2.4 MB.
    float* dinv = (float*)d_ws;
    float* A    = dinv + N;
    float* g    = A + N;
    float* out  = (float*)d_out;

    const int TB = 256;
    const int nb = (N + TB - 1) / TB;
    const int wb = (N + 127) / 128;      // 8 waves/block * 16 nodes/wave

    k_init_deg<<<nb, TB, 0, stream>>>((unsigned*)dinv, N);

    const bool vec4 = (E % 4 == 0);      // keeps dst4 pointer 16B-aligned
    if (vec4) {
        const int E4  = E / 4;
        const int eb4 = (E4 + TB - 1) / TB;
        const int4* src4 = (const int4*)src;
        const int4* dst4 = (const int4*)dst;

        k_count_deg4<<<eb4, TB, 0, stream>>>(dst4, (unsigned*)dinv, E4);
        k_dinv_selfloop<<<nb, TB, 0, stream>>>((unsigned*)dinv, dinv, x, A, N);
        k_scatter4<<<eb4, TB, 0, stream>>>(src4, dst4, dinv, x, /*stride=*/4, A, E4);
        k_transform_wmma<<<wb, TB, 0, stream>>>(A, W1, b1, W2, g, N);
        k_out_init<<<nb, TB, 0, stream>>>(dinv, g, b2, out, N);
        k_scatter4<<<eb4, TB, 0, stream>>>(src4, dst4, dinv, g, /*stride=*/1, out, E4);
    } else {
        const int eb = (E + TB - 1) / TB;
        k_count_deg<<<eb, TB, 0, stream>>>(dst, (unsigned*)dinv, E);
        k_dinv_selfloop<<<nb, TB, 0, stream>>>((unsigned*)dinv, dinv, x, A, N);
        k_scatter<<<eb, TB, 0, stream>>>(src, dst, dinv, x, /*stride=*/4, A, E);
        k_transform_wmma<<<wb, TB, 0, stream>>>(A, W1, b1, W2, g, N);
        k_out_init<<<nb, TB, 0, stream>>>(dinv, g, b2, out, N);
        k_scatter<<<eb, TB, 0, stream>>>(src, dst, dinv, g, /*stride=*/1, out, E);
    }
}